// MultiResolutionHashEncoding_78975858639607
// MI455X (gfx1250) — compile-verified
//
#include <hip/hip_runtime.h>
#include <hip/hip_bf16.h>

// Multi-resolution hash encoding (Instant-NGP style), MI455X / gfx1250.
//
// Mapping: one lane = one (point, level) pair; 16 levels per point, so a
// wave32 covers exactly 2 points. Output float2 stores from lanes 0..15 land
// on out[n*32 .. n*32+31] (contiguous 128B) and lanes 16..31 on the next
// point's 128B -> fully coalesced b64 stores.
//
// Memory policy (reasoned from 192MB L2, 23.3 TB/s HBM):
//  - hash table (64MB) fits in L2 -> default RT policy, keep it resident.
//  - x (6MB, read-once) -> non-temporal load (global_load_b96 th:TH_LOAD_NT).
//  - out (64MB, write-once) -> non-temporal store (don't rinse the table
//    out of L2).
// All 8 corner gathers are issued back-to-back (independent) so the wave has
// 8 loads in flight before the single wait; 8 waves/block for latency hiding.
//
// The power-of-two table_size check is wave-uniform (scalar load), so it is
// written as a real if/else around the whole gather+blend body to get an
// s_cbranch — the executed path is pure v_and + global_load_b64, with the
// integer-modulo fallback kept in cold code only.

typedef float f32x2 __attribute__((ext_vector_type(2)));

#define NLEV 16

__device__ __forceinline__ void gather_blend(
    const f32x2* __restrict__ table2,
    const unsigned* h, const float* w, unsigned off,
    float& a0, float& a1,
    unsigned msk, unsigned ts, bool use_mask)
{
    f32x2 f[8];
#pragma unroll
    for (int c = 0; c < 8; ++c) {
        unsigned hm = use_mask ? (h[c] & msk) : (h[c] % ts);
        unsigned idx = hm + off;          // < 8.4M -> 32-bit GVS addressing
        f[c] = table2[idx];
    }
#pragma unroll
    for (int c = 0; c < 8; ++c) {
        a0 = fmaf(w[c], f[c].x, a0);
        a1 = fmaf(w[c], f[c].y, a1);
    }
}

__global__ __launch_bounds__(256) void
MultiResolutionHashEncoding_78975858639607_kernel(
    const float* __restrict__ x,            // [N,3]
    const float* __restrict__ table,        // [TS*L, 2]
    const float* __restrict__ scalings,     // [L]
    const int*   __restrict__ hash_offset,  // [L]
    const int*   __restrict__ table_size_p, // scalar
    float*       __restrict__ out,          // [N, L*2]
    int n_points)
{
    const int tid = blockIdx.x * blockDim.x + threadIdx.x;
    const int l = tid & (NLEV - 1);
    const int n = tid >> 4;
    if (n >= n_points) return;

    const unsigned ts   = (unsigned)*table_size_p;   // uniform scalar load
    const unsigned msk  = ts - 1u;
    const bool     pow2 = (ts & msk) == 0u;          // uniform condition

    const float    s   = scalings[l];
    const unsigned off = (unsigned)hash_offset[l];

    // read-once input: stream past L2 (non-temporal b96)
    const float px = __builtin_nontemporal_load(x + 3 * (size_t)n + 0);
    const float py = __builtin_nontemporal_load(x + 3 * (size_t)n + 1);
    const float pz = __builtin_nontemporal_load(x + 3 * (size_t)n + 2);

    const float sx = px * s, sy = py * s, sz = pz * s;
    const float fx = floorf(sx), fy = floorf(sy), fz = floorf(sz);
    const float gx = ceilf(sx),  gy = ceilf(sy),  gz = ceilf(sz);
    const float tx = sx - fx, ty = sy - fy, tz = sz - fz;
    const float ux = 1.0f - tx, uy = 1.0f - ty, uz = 1.0f - tz;

    // NGP hash components (uint32 wraparound multiply, as in the reference)
    const unsigned P1 = 2654435761u, P2 = 805459861u;
    const unsigned hx0 = (unsigned)(int)fx;          // prime 1 for x-axis
    const unsigned hx1 = (unsigned)(int)gx;
    const unsigned hy0 = (unsigned)(int)fy * P1;
    const unsigned hy1 = (unsigned)(int)gy * P1;
    const unsigned hz0 = (unsigned)(int)fz * P2;
    const unsigned hz1 = (unsigned)(int)gz * P2;

    // Corner order v0..v7 per reference _CORNER_SIGNS (1=ceil, 0=floor)
    unsigned h[8];
    h[0] = hx1 ^ hy1 ^ hz1;   // (c,c,c)
    h[1] = hx1 ^ hy1 ^ hz0;   // (c,c,f)
    h[2] = hx1 ^ hy0 ^ hz1;   // (c,f,c)
    h[3] = hx0 ^ hy1 ^ hz1;   // (f,c,c)
    h[4] = hx1 ^ hy0 ^ hz0;   // (c,f,f)
    h[5] = hx0 ^ hy1 ^ hz0;   // (f,c,f)
    h[6] = hx0 ^ hy0 ^ hz1;   // (f,f,c)
    h[7] = hx0 ^ hy0 ^ hz0;   // (f,f,f)

    // Weights exactly as the reference's (quirky) interpolation tree
    float w[8];
    w[0] = tx * ty * tz;
    w[1] = tx * uy * tz;
    w[2] = ux * uy * tz;
    w[3] = ux * ty * tz;
    w[4] = tx * ty * uz;
    w[5] = tx * uy * uz;
    w[6] = ux * uy * uz;
    w[7] = ux * ty * uz;

    const f32x2* __restrict__ table2 = (const f32x2*)table;
    float a0 = 0.0f, a1 = 0.0f;

    if (pow2) {
        // hot path: v_and + 8x global_load_b64, table L2-resident (RT policy)
        gather_blend(table2, h, w, off, a0, a1, msk, ts, true);
    } else {
        // cold generic-modulo fallback (never taken for TS = 2^19)
        gather_blend(table2, h, w, off, a0, a1, msk, ts, false);
    }

    // write-once output: non-temporal b64 store, coalesced per wave
    f32x2 r; r.x = a0; r.y = a1;
    __builtin_nontemporal_store(
        r, (f32x2*)(out + (size_t)n * (NLEV * 2) + l * 2));
}

extern "C" void kernel_launch(void* const* d_in, const int* in_sizes, int n_in,
                              void* d_out, int out_size, void* d_ws, size_t ws_size,
                              hipStream_t stream) {
    (void)n_in; (void)out_size; (void)d_ws; (void)ws_size;
    const float* x        = (const float*)d_in[0];
    const float* table    = (const float*)d_in[1];
    const float* scalings = (const float*)d_in[2];
    const int*   hoff     = (const int*)d_in[3];
    const int*   ts       = (const int*)d_in[4];
    float*       out      = (float*)d_out;

    const int n_points = in_sizes[0] / 3;          // x is [N,3]
    const long total   = (long)n_points * NLEV;    // one lane per (point,level)
    const int  block   = 256;                      // 8 wave32 per block
    const long grid    = (total + block - 1) / block;

    MultiResolutionHashEncoding_78975858639607_kernel
        <<<dim3((unsigned)grid), dim3(block), 0, stream>>>(
            x, table, scalings, hoff, ts, out, n_points);
}